// DetectorEndpoint_29180007809727
// MI455X (gfx1250) — compile-verified
//
#include <hip/hip_runtime.h>
#include <hip/hip_bf16.h>

#define NB 8
#define NA 49104
#define NC 80
#define KPRE 200
#define KPC 100
#define MAXDET 100
#define NCAND (NC * KPC) /* 8000 */
#define TILES 13
#define PADN (TILES * 16) /* 208 */

typedef float v2f __attribute__((ext_vector_type(2)));
typedef float v8f __attribute__((ext_vector_type(8)));

#ifndef __has_builtin
#define __has_builtin(x) 0
#endif
#if __has_builtin(__builtin_amdgcn_wmma_f32_16x16x4_f32)
#define HAVE_WMMA4 1
#else
#define HAVE_WMMA4 0
#endif

__device__ __forceinline__ unsigned fkey(float f) {
  unsigned u = __float_as_uint(f);
  return (u & 0x80000000u) ? ~u : (u | 0x80000000u);
}
__device__ __forceinline__ float kinv(unsigned k) {
  unsigned u = (k & 0x80000000u) ? (k & 0x7FFFFFFFu) : ~k;
  return __uint_as_float(u);
}

// ---------------- Kernel 1: box decode (anchors computed analytically) ----------------
__global__ __launch_bounds__(256) void decode_kernel(const float* __restrict__ pred,
                                                     float4* __restrict__ boxes) {
  int gid = blockIdx.x * blockDim.x + threadIdx.x;
  if (gid >= NB * NA) return;
  int a = gid % NA;

  int lvl, rel;
  if (a < 36864)      { lvl = 3; rel = a; }
  else if (a < 46080) { lvl = 4; rel = a - 36864; }
  else if (a < 48384) { lvl = 5; rel = a - 46080; }
  else if (a < 48960) { lvl = 6; rel = a - 48384; }
  else                { lvl = 7; rel = a - 48960; }
  int fw = 512 >> lvl;
  float stride = (float)(1 << lvl);
  int cell = rel / 9, k = rel - cell * 9;
  int x = cell % fw, y = cell / fw;
  float acx = (x + 0.5f) * stride;
  float acy = (y + 0.5f) * stride;
  float side = (float)(4 << lvl);
  float area = side * side;
  const float ratios[3] = {0.5f, 1.0f, 2.0f};
  const float scales[3] = {1.0f, 1.2599210498948732f, 1.5874010519681994f};
  float r = ratios[k / 3], s = scales[k % 3];
  float ah = sqrtf(area / r);
  float aw = area / ah;
  float dw = s * aw, dh = s * ah;

  const float4 p = ((const float4*)pred)[(size_t)gid * 21];  // first 4 of 84 floats
  float cx = p.x * 0.1f * dw + acx;
  float cy = p.y * 0.1f * dh + acy;
  float w  = expf(p.z * 0.2f) * dw;
  float h  = expf(p.w * 0.2f) * dh;
  boxes[gid] = make_float4(cx - 0.5f * w, cy - 0.5f * h, cx + 0.5f * w, cy + 0.5f * h);
}

// ------------- Kernel 2: per-(batch,class) top-200 select + NMS + compact -------------
__global__ __launch_bounds__(256) void nms_class_kernel(const float* __restrict__ pred,
                                                        const float4* __restrict__ boxes,
                                                        float* __restrict__ cand_score,
                                                        float4* __restrict__ cand_box) {
  __shared__ unsigned s_hist[256];
  __shared__ unsigned s_prefix, s_rem, s_cnt, s_eq;
  __shared__ unsigned s_key[256];
  __shared__ int s_idx[256];
  __shared__ float s_x0[PADN], s_y0[PADN], s_x1[PADN], s_y1[PADN], s_area[PADN];
  __shared__ unsigned s_mask[PADN * 7];
  __shared__ unsigned s_keep[7];
  __shared__ float s_oscore[KPC];
  __shared__ float4 s_obox[KPC];

  const int tid = threadIdx.x;
  const int b = blockIdx.x / NC, c = blockIdx.x % NC;
  const float* sc = pred + (size_t)b * NA * 84 + 4 + c;

  if (tid == 0) { s_prefix = 0u; s_rem = KPRE; s_cnt = 0u; s_eq = 0u; }

  // ---- exact 200th-largest via 4x8-bit radix select (logit keys; sigmoid is monotone)
  for (int shift = 24; shift >= 0; shift -= 8) {
    s_hist[tid] = 0u;
    __syncthreads();
    unsigned pfx = s_prefix;
    for (int a = tid; a < NA; a += 256) {
      unsigned key = fkey(sc[(size_t)a * 84]);
      bool match = (shift == 24) || ((key >> (shift + 8)) == (pfx >> (shift + 8)));
      if (match) atomicAdd(&s_hist[(key >> shift) & 255], 1u);
    }
    __syncthreads();
    if (tid == 0) {
      unsigned rem = s_rem;
      for (int bin = 255; bin >= 0; --bin) {
        unsigned cnt = s_hist[bin];
        if (cnt >= rem) { s_prefix = pfx | ((unsigned)bin << shift); break; }
        rem -= cnt;
      }
      s_rem = rem;
    }
    __syncthreads();
  }

  // ---- collect exactly 200 candidates
  {
    unsigned thr = s_prefix, rem = s_rem;
    for (int a = tid; a < NA; a += 256) {
      unsigned key = fkey(sc[(size_t)a * 84]);
      if (key > thr) {
        unsigned p = atomicAdd(&s_cnt, 1u);
        s_key[p] = key; s_idx[p] = a;
      } else if (key == thr) {
        unsigned e = atomicAdd(&s_eq, 1u);
        if (e < rem) { unsigned p = atomicAdd(&s_cnt, 1u); s_key[p] = key; s_idx[p] = a; }
      }
    }
  }
  __syncthreads();
  if (tid >= KPRE) { s_key[tid] = 0u; s_idx[tid] = 1000000 + tid; }
  __syncthreads();

  // ---- bitonic sort 256: key desc, index asc (top_k tie-break)
  for (int k = 2; k <= 256; k <<= 1) {
    for (int j = k >> 1; j > 0; j >>= 1) {
      __syncthreads();
      int i = tid, l = i ^ j;
      if (l > i) {
        unsigned ki = s_key[i], kl = s_key[l];
        int ii = s_idx[i], il = s_idx[l];
        bool gi = (ki > kl) || (ki == kl && ii < il);
        bool desc = ((i & k) == 0);
        if (desc ? !gi : gi) { s_key[i] = kl; s_key[l] = ki; s_idx[i] = il; s_idx[l] = ii; }
      }
    }
  }
  __syncthreads();

  // ---- stage boxes + areas, zero masks
  if (tid < PADN) {
    if (tid < KPRE) {
      float4 bx = boxes[(size_t)b * NA + s_idx[tid]];
      s_x0[tid] = bx.x; s_y0[tid] = bx.y; s_x1[tid] = bx.z; s_y1[tid] = bx.w;
      s_area[tid] = (bx.z - bx.x) * (bx.w - bx.y);
    } else {
      s_x0[tid] = 0.f; s_y0[tid] = 0.f; s_x1[tid] = 0.f; s_y1[tid] = 0.f; s_area[tid] = 0.f;
    }
  }
  for (int m = tid; m < PADN * 7; m += 256) s_mask[m] = 0u;
  if (tid < 7) s_keep[tid] = 0u;
  __syncthreads();

  // ---- 13x13 tiles of 16x16: union-sum via WMMA f32 16x16x4 (rank-2 matmul),
  //      intersection via VALU, pack iou>0.5 bits
  {
    int lane = tid & 31, wave = tid >> 5;
    int l15 = lane & 15;
    bool lo = lane < 16;
    for (int t = wave; t < TILES * TILES; t += 8) {
      int Mb = (t / TILES) * 16, Nb = (t % TILES) * 16;
      int N = Nb + l15;
      int Madd = lo ? 0 : 8;
#if HAVE_WMMA4
      // A(16x4): A[M][0]=area_i[M], A[M][1]=1 ; lanes0-15 hold K={0,1}, lanes16-31 K={2,3}=0
      // B(4x16): B[0][N]=1, B[1][N]=area_j[N] ; VGPR0 lanes0-15=K0, hi=K2; VGPR1 lo=K1, hi=K3
      v2f av, bv;
      av.x = lo ? s_area[Mb + l15] : 0.0f;
      av.y = lo ? 1.0f : 0.0f;
      bv.x = lo ? 1.0f : 0.0f;
      bv.y = lo ? s_area[N] : 0.0f;
      v8f cv = {0.f, 0.f, 0.f, 0.f, 0.f, 0.f, 0.f, 0.f};
      cv = __builtin_amdgcn_wmma_f32_16x16x4_f32(false, av, false, bv, (short)0, cv,
                                                 false, false);
#endif
      float jx0 = s_x0[N], jy0 = s_y0[N], jx1 = s_x1[N], jy1 = s_y1[N];
#pragma unroll
      for (int v = 0; v < 8; ++v) {
        int M = Mb + Madd + v;  // C/D layout: lanes0-15 -> M=v, lanes16-31 -> M=v+8
#if HAVE_WMMA4
        float asum = cv[v];
#else
        float asum = s_area[M] + s_area[N];
#endif
        float ltx = fmaxf(s_x0[M], jx0), lty = fmaxf(s_y0[M], jy0);
        float rbx = fminf(s_x1[M], jx1), rby = fminf(s_y1[M], jy1);
        float inter = fmaxf(rbx - ltx, 0.0f) * fmaxf(rby - lty, 0.0f);
        float uni = fmaxf(asum - inter, 1e-8f);
        if (inter > 0.5f * uni)  // iou > 0.5
          atomicOr(&s_mask[M * 7 + (N >> 5)], 1u << (N & 31));
      }
    }
  }
  __syncthreads();

  // ---- keep0 = score > 0.5  <=>  logit key > key(+0.0)
  if (tid < KPRE && s_key[tid] > 0x80000000u) atomicOr(&s_keep[tid >> 5], 1u << (tid & 31));
  __syncthreads();

  // ---- greedy suppression sweep (sequential; trivial work on bitmask)
  if (tid == 0) {
    for (int i = 0; i < KPRE; ++i) {
      if (!((s_keep[i >> 5] >> (i & 31)) & 1u)) continue;
      int wi = i >> 5, bi = i & 31;
      for (int w = wi; w < 7; ++w) {
        unsigned m = s_mask[i * 7 + w];
        if (w == wi) m &= (bi == 31) ? 0u : (0xFFFFFFFFu << (bi + 1));  // only j > i
        s_keep[w] &= ~m;
      }
    }
  }
  __syncthreads();

  // ---- compact kept (already score-sorted) into first <=100 slots
  if (tid < KPC) { s_oscore[tid] = 0.0f; s_obox[tid] = make_float4(0.f, 0.f, 0.f, 0.f); }
  __syncthreads();
  if (tid < KPRE) {
    unsigned wi = tid >> 5, bi = tid & 31;
    if ((s_keep[wi] >> bi) & 1u) {
      int rank = __popc(s_keep[wi] & ((1u << bi) - 1u));
      for (unsigned w = 0; w < wi; ++w) rank += __popc(s_keep[w]);
      if (rank < KPC) {
        float logit = kinv(s_key[tid]);
        s_oscore[rank] = 1.0f / (1.0f + expf(-logit));
        s_obox[rank] = make_float4(s_x0[tid], s_y0[tid], s_x1[tid], s_y1[tid]);
      }
    }
  }
  __syncthreads();
  if (tid < KPC) {
    size_t o = (size_t)b * NCAND + (size_t)c * KPC + tid;
    cand_score[o] = s_oscore[tid];
    cand_box[o] = s_obox[tid];
  }
}

// ---------------- Kernel 3: per-batch top-100 over 8000 candidates -------------------
__global__ __launch_bounds__(256) void final_topk_kernel(const float* __restrict__ cand_score,
                                                         const float4* __restrict__ cand_box,
                                                         float* __restrict__ out) {
  __shared__ unsigned s_hist[256];
  __shared__ unsigned s_prefix, s_rem, s_cnt, s_eq;
  __shared__ unsigned s_key[128];
  __shared__ int s_idx[128];
  __shared__ int s_valid;

  const int tid = threadIdx.x, b = blockIdx.x;
  const float* scp = cand_score + (size_t)b * NCAND;
  if (tid == 0) { s_prefix = 0u; s_rem = MAXDET; s_cnt = 0u; s_eq = 0u; s_valid = 0; }

  for (int shift = 24; shift >= 0; shift -= 8) {
    s_hist[tid] = 0u;
    __syncthreads();
    unsigned pfx = s_prefix;
    for (int a = tid; a < NCAND; a += 256) {
      unsigned key = fkey(scp[a]);
      bool match = (shift == 24) || ((key >> (shift + 8)) == (pfx >> (shift + 8)));
      if (match) atomicAdd(&s_hist[(key >> shift) & 255], 1u);
    }
    __syncthreads();
    if (tid == 0) {
      unsigned rem = s_rem;
      for (int bin = 255; bin >= 0; --bin) {
        unsigned cnt = s_hist[bin];
        if (cnt >= rem) { s_prefix = pfx | ((unsigned)bin << shift); break; }
        rem -= cnt;
      }
      s_rem = rem;
    }
    __syncthreads();
  }
  {
    unsigned thr = s_prefix, rem = s_rem;
    for (int a = tid; a < NCAND; a += 256) {
      unsigned key = fkey(scp[a]);
      if (key > thr) {
        unsigned p = atomicAdd(&s_cnt, 1u);
        s_key[p] = key; s_idx[p] = a;
      } else if (key == thr) {
        unsigned e = atomicAdd(&s_eq, 1u);
        if (e < rem) { unsigned p = atomicAdd(&s_cnt, 1u); s_key[p] = key; s_idx[p] = a; }
      }
    }
  }
  __syncthreads();
  if (tid >= MAXDET && tid < 128) { s_key[tid] = 0u; s_idx[tid] = NCAND + tid; }
  __syncthreads();

  for (int k = 2; k <= 128; k <<= 1) {
    for (int j = k >> 1; j > 0; j >>= 1) {
      __syncthreads();
      if (tid < 128) {
        int i = tid, l = i ^ j;
        if (l > i) {
          unsigned ki = s_key[i], kl = s_key[l];
          int ii = s_idx[i], il = s_idx[l];
          bool gi = (ki > kl) || (ki == kl && ii < il);
          bool desc = ((i & k) == 0);
          if (desc ? !gi : gi) { s_key[i] = kl; s_key[l] = ki; s_idx[i] = il; s_idx[l] = ii; }
        }
      }
    }
  }
  __syncthreads();

  if (tid < MAXDET) {
    float s = kinv(s_key[tid]);
    bool ok = s > 0.5f;
    float4 bx = make_float4(0.f, 0.f, 0.f, 0.f);
    float cl = 0.f, sv = 0.f;
    if (ok) {
      int idx = s_idx[tid];
      bx = cand_box[(size_t)b * NCAND + idx];
      cl = (float)(idx / KPC);
      sv = s;
      atomicAdd(&s_valid, 1);
    }
    size_t ob = (size_t)b * 400 + (size_t)tid * 4;
    out[ob + 0] = bx.x; out[ob + 1] = bx.y; out[ob + 2] = bx.z; out[ob + 3] = bx.w;
    out[3200 + b * 100 + tid] = sv;
    out[4000 + b * 100 + tid] = cl;
  }
  __syncthreads();
  if (tid == 0) out[4800 + b] = (float)s_valid;
}

extern "C" void kernel_launch(void* const* d_in, const int* in_sizes, int n_in,
                              void* d_out, int out_size, void* d_ws, size_t ws_size,
                              hipStream_t stream) {
  (void)in_sizes; (void)n_in; (void)out_size; (void)ws_size;
  const float* pred = (const float*)d_in[1];  // d_in[0] = image (only shape used)

  float* boxes = (float*)d_ws;                                   // NB*NA*4 floats (6.3 MB)
  float* cand_score = boxes + (size_t)NB * NA * 4;               // NB*8000 floats
  float4* cand_box = (float4*)(cand_score + (size_t)NB * NCAND); // NB*8000 float4

  decode_kernel<<<(NB * NA + 255) / 256, 256, 0, stream>>>(pred, (float4*)boxes);
  nms_class_kernel<<<NB * NC, 256, 0, stream>>>(pred, (const float4*)boxes,
                                                cand_score, cand_box);
  final_topk_kernel<<<NB, 256, 0, stream>>>(cand_score, cand_box, (float*)d_out);
}